// _ScaledDotProductAttention_13984413515924
// MI455X (gfx1250) — compile-verified
//
#include <hip/hip_runtime.h>

typedef float v2f __attribute__((ext_vector_type(2)));
typedef float v8f __attribute__((ext_vector_type(8)));

#define B_ 2
#define H_ 16
#define S_ 2048
#define D_ 64
#define SCALE 0.125f /* 64^-0.5 */

// One workgroup = 16 rows of one (b,h). 8 waves; wave w owns columns
// [256w, 256w+256). WMMA shape: f32 16x16x4 (native fp32 path — this problem
// is HBM-bound, so we keep exact reference numerics instead of downcasting).
__launch_bounds__(256)
__global__ void attn_bias_softmax_kernel(const float* __restrict__ q,
                                         const float* __restrict__ k,   // (B,H,D,S) pre-transposed
                                         const float* __restrict__ v,
                                         const float* __restrict__ prev,
                                         float* __restrict__ out,       // (B,H,S,D)
                                         float* __restrict__ weights,   // (B,H,S,S)
                                         float* __restrict__ scores)    // (B,H,S,S)
{
    __shared__ float red[16][8];        // per-row, per-wave partials
    __shared__ float rowstat[16];       // broadcast row max / inv-sum
    __shared__ float wstage[8][16][17]; // per-wave 16x16 weight tile (padded)
    __shared__ float outacc[16][65];    // cross-wave output reduction (padded)

    const int tid  = threadIdx.x;
    const int wave = tid >> 5;
    const int lane = tid & 31;
    const int half = lane >> 4;   // wave32: lanes 0-15 vs 16-31
    const int l16  = lane & 15;

    const int bh   = blockIdx.x >> 7;   // 128 row-blocks per (b,h)
    const int rb   = blockIdx.x & 127;
    const int row0 = rb << 4;

    const float* qp = q    + (size_t)bh * S_ * D_ + (size_t)row0 * D_;
    const float* kp = k    + (size_t)bh * D_ * S_;
    const float* vp = v    + (size_t)bh * S_ * D_;
    const float* pp = prev + (size_t)bh * S_ * S_ + (size_t)row0 * S_;
    float* sp = scores  + (size_t)bh * S_ * S_ + (size_t)row0 * S_;
    float* wp = weights + (size_t)bh * S_ * S_ + (size_t)row0 * S_;
    float* op = out     + (size_t)bh * S_ * D_ + (size_t)row0 * D_;

    // zero the cross-wave output accumulator
    for (int i = tid; i < 16 * 64; i += 256)
        outacc[i >> 6][i & 63] = 0.0f;

    // ---- Q A-fragments: 16x4 f32 A-layout, 16 k-steps over D=64 -----------
    // lane (half,l16): row = l16, holds K = 4*kk + 2*half + {0,1}
    v2f aq[16];
#pragma unroll
    for (int kk = 0; kk < 16; ++kk) {
        const int kb = kk * 4 + half * 2;
        aq[kk] = *(const v2f*)(qp + l16 * D_ + kb);
    }

    const int colw = wave << 8;     // this wave's 256-column window

    // ---- phase 1: scores = Q*K^T*SCALE + prev; stream out; row max --------
    float rmax[8];
#pragma unroll
    for (int i = 0; i < 8; ++i) rmax[i] = -__builtin_inff();

    for (int t = 0; t < 16; ++t) {
        const int col0 = colw + t * 16;
        v8f acc = {};
#pragma unroll
        for (int kk = 0; kk < 16; ++kk) {
            const int krow = kk * 4 + half * 2;
            v2f b;
            b.x = kp[(size_t)krow * S_ + col0 + l16];
            b.y = kp[(size_t)(krow + 1) * S_ + col0 + l16];
            acc = __builtin_amdgcn_wmma_f32_16x16x4_f32(false, aq[kk], false, b,
                                                        (short)0, acc, false, false);
        }
        // C-layout: VGPR vi -> row vi (lanes 0-15) / vi+8 (lanes 16-31), col = l16
#pragma unroll
        for (int vi = 0; vi < 8; ++vi) {
            const int r = vi + half * 8;
            const int c = col0 + l16;
            float s = acc[vi] * SCALE + pp[(size_t)r * S_ + c];
            sp[(size_t)r * S_ + c] = s;
            rmax[vi] = fmaxf(rmax[vi], s);
        }
    }
    // reduce max over the 16 columns held by this half-wave (xor 1..8 stays in-half)
#pragma unroll
    for (int vi = 0; vi < 8; ++vi)
#pragma unroll
        for (int off = 1; off < 16; off <<= 1)
            rmax[vi] = fmaxf(rmax[vi], __shfl_xor(rmax[vi], off, 32));
    if (l16 == 0)
#pragma unroll
        for (int vi = 0; vi < 8; ++vi) red[vi + half * 8][wave] = rmax[vi];
    __syncthreads();
    if (tid < 16) {
        float m = red[tid][0];
#pragma unroll
        for (int w = 1; w < 8; ++w) m = fmaxf(m, red[tid][w]);
        rowstat[tid] = m;
    }
    __syncthreads();
    float gmax[8];
#pragma unroll
    for (int vi = 0; vi < 8; ++vi) gmax[vi] = rowstat[vi + half * 8];

    // ---- phase 2: e = exp(s - max); unnormalised weights; row sums --------
    // scores re-read is same-wave same-address (in-order per ISA) and L2-hot.
    float rsum[8];
#pragma unroll
    for (int i = 0; i < 8; ++i) rsum[i] = 0.0f;

    for (int t = 0; t < 16; ++t) {
        const int col0 = colw + t * 16;
#pragma unroll
        for (int vi = 0; vi < 8; ++vi) {
            const int r = vi + half * 8;
            const int c = col0 + l16;
            float e = __expf(sp[(size_t)r * S_ + c] - gmax[vi]);
            wp[(size_t)r * S_ + c] = e;
            rsum[vi] += e;
        }
    }
#pragma unroll
    for (int vi = 0; vi < 8; ++vi)
#pragma unroll
        for (int off = 1; off < 16; off <<= 1)
            rsum[vi] += __shfl_xor(rsum[vi], off, 32);
    __syncthreads();   // gmax reads done before rowstat/red reuse
    if (l16 == 0)
#pragma unroll
        for (int vi = 0; vi < 8; ++vi) red[vi + half * 8][wave] = rsum[vi];
    __syncthreads();
    if (tid < 16) {
        float s = 0.0f;
#pragma unroll
        for (int w = 0; w < 8; ++w) s += red[tid][w];
        rowstat[tid] = 1.0f / s;
    }
    __syncthreads();
    float rinv[8];
#pragma unroll
    for (int vi = 0; vi < 8; ++vi) rinv[vi] = rowstat[vi + half * 8];

    // ---- phase 3: normalise weights; out = W @ V via WMMA ------------------
    v8f oacc[4];
#pragma unroll
    for (int n = 0; n < 4; ++n) oacc[n] = (v8f){};

    for (int t = 0; t < 16; ++t) {
        const int col0 = colw + t * 16;
        // normalise this 16x16 tile, write final weights, stage into LDS
#pragma unroll
        for (int vi = 0; vi < 8; ++vi) {
            const int r = vi + half * 8;
            const int c = col0 + l16;
            float wgt = wp[(size_t)r * S_ + c] * rinv[vi];
            wp[(size_t)r * S_ + c] = wgt;
            wstage[wave][r][l16] = wgt;   // per-wave slice; DS ops in-order per wave
        }
        // 4 k-steps of 4 over the tile's 16 sequence positions
#pragma unroll
        for (int j = 0; j < 4; ++j) {
            const int kb = j * 4 + half * 2;
            v2f a;
            a.x = wstage[wave][l16][kb];
            a.y = wstage[wave][l16][kb + 1];
            const int kg = col0 + j * 4 + half * 2;
#pragma unroll
            for (int n = 0; n < 4; ++n) {
                v2f b;
                b.x = vp[(size_t)kg * D_ + n * 16 + l16];
                b.y = vp[(size_t)(kg + 1) * D_ + n * 16 + l16];
                oacc[n] = __builtin_amdgcn_wmma_f32_16x16x4_f32(false, a, false, b,
                                                                (short)0, oacc[n],
                                                                false, false);
            }
        }
    }
    // cross-wave reduction: ds_add_f32 atomics into LDS
#pragma unroll
    for (int n = 0; n < 4; ++n)
#pragma unroll
        for (int vi = 0; vi < 8; ++vi)
            atomicAdd(&outacc[vi + half * 8][n * 16 + l16], oacc[n][vi]);
    __syncthreads();

    for (int i = tid; i < 16 * 64; i += 256)
        op[(size_t)(i >> 6) * D_ + (i & 63)] = outacc[i >> 6][i & 63];
}

extern "C" void kernel_launch(void* const* d_in, const int* in_sizes, int n_in,
                              void* d_out, int out_size, void* d_ws, size_t ws_size,
                              hipStream_t stream) {
    const float* q    = (const float*)d_in[0];
    const float* k    = (const float*)d_in[1];
    const float* v    = (const float*)d_in[2];
    const float* prev = (const float*)d_in[3];

    float* out     = (float*)d_out;                         // (B,H,S,D)
    float* weights = out + (size_t)B_ * H_ * S_ * D_;       // (B,H,S,S)
    float* scores  = weights + (size_t)B_ * H_ * S_ * S_;   // (B,H,S,S)

    dim3 grid(B_ * H_ * (S_ / 16));   // 4096 workgroups
    dim3 block(256);                  // 8 wave32 waves
    attn_bias_softmax_kernel<<<grid, block, 0, stream>>>(q, k, v, prev,
                                                         out, weights, scores);
}